// SparseAutoencoder_79173427134760
// MI455X (gfx1250) — compile-verified
//
#include <hip/hip_runtime.h>

typedef __attribute__((ext_vector_type(2))) float v2f;
typedef __attribute__((ext_vector_type(8))) float v8f;
typedef __attribute__((ext_vector_type(4))) unsigned u32x4;
typedef __attribute__((ext_vector_type(8))) unsigned u32x8;

#define D_MODEL 768
#define D_DICT  49152
#define BATCH   4096
#define TOPK    500
#define KCAP    512

// encoder GEMM tiling
#define BM 64
#define BN 128
#define TK 32
#define KSTAGES (D_MODEL / TK)   // 24
#define PAD_ROW 33               // TK+1 row pitch, produced by TDM LDS padding

// ---------------------------------------------------------------------------
// LDS byte offset of a __shared__ object (generic -> AS(3) addrspacecast).
// Needed to embed the LDS destination in a TDM descriptor.
// ---------------------------------------------------------------------------
__device__ __forceinline__ unsigned lds_byte_off(const void* p) {
    return (unsigned)(uintptr_t)(__attribute__((address_space(3))) const void*)p;
}

// ---------------------------------------------------------------------------
// Issue one 2-D TENSOR_LOAD_TO_LDS via Tensor Data Mover (ISA ch.8 D# layout).
//  - data_size = 4 bytes, tile = tile_d0 x tile_d1, tensor row stride = stride0
//  - pad_enable: +1 dword after every 32 dwords => LDS row pitch 33 floats
//  - groups 2/3 zeroed (2-D tile: tile_dim2..4 = 0, iterate off)
// Tracked by TENSORcnt; drained with s_wait_tensorcnt.
// ---------------------------------------------------------------------------
__device__ __forceinline__ void tdm_load_2d(unsigned lds_off, unsigned long long gaddr,
                                            unsigned tensor_d0, unsigned tensor_d1,
                                            unsigned tile_d0, unsigned tile_d1,
                                            unsigned stride0) {
    u32x4 g0;
    g0[0] = 1u;                                            // count=1 (valid user D#)
    g0[1] = lds_off;                                       // lds_addr (bytes)
    g0[2] = (unsigned)(gaddr & 0xffffffffull);             // global_addr[31:0]
    g0[3] = (unsigned)((gaddr >> 32) & 0x01ffffffull)      // global_addr[56:32]
          | 0x80000000u;                                   // type=2 ("image")
    u32x8 g1;
    g1[0] = (2u << 16)     // data_size = 4B
          | (1u << 20)     // pad_enable
          | (4u << 22);    // pad_interval code 4 = 32 dwords; pad_amount code 0 = 1 dword
    g1[1] = (tensor_d0 & 0xffffu) << 16;                   // tensor_dim0[15:0]
    g1[2] = (tensor_d0 >> 16) | ((tensor_d1 & 0xffffu) << 16);
    g1[3] = (tensor_d1 >> 16) | (tile_d0 << 16);           // tile_dim0
    g1[4] = tile_d1;                                       // tile_dim1 (tile_dim2=0)
    g1[5] = stride0;                                       // tensor_dim0_stride[31:0]
    g1[6] = 0u;
    g1[7] = 0u;
    const u32x4 z4 = {0u, 0u, 0u, 0u};
    asm volatile("tensor_load_to_lds %0, %1, %2, %3"
                 :: "s"(g0), "s"(g1), "s"(z4), "s"(z4)
                 : "memory");
}

// ---------------------------------------------------------------------------
// Kernel 1: transpose W_dec [768 x 49152] -> WdecT [49152 x 768] (workspace).
// ---------------------------------------------------------------------------
__global__ void __launch_bounds__(256)
k_transpose(const float* __restrict__ Wdec, float* __restrict__ WdecT) {
    __shared__ float tile[32][33];
    const int tx = threadIdx.x, ty = threadIdx.y;      // 32 x 8
    const int d0 = blockIdx.x * 32, m0 = blockIdx.y * 32;
#pragma unroll
    for (int k = 0; k < 4; ++k) {
        int m = m0 + ty + 8 * k;
        tile[ty + 8 * k][tx] = Wdec[(size_t)m * D_DICT + (d0 + tx)];
    }
    __syncthreads();
#pragma unroll
    for (int k = 0; k < 4; ++k) {
        int d = d0 + ty + 8 * k;
        WdecT[(size_t)d * D_MODEL + (m0 + tx)] = tile[tx][ty + 8 * k];
    }
}

// ---------------------------------------------------------------------------
// Kernel 2: encoder GEMM  post = relu(x @ W_enc^T + b_enc)  [M=4096,N=49152,K=768]
// Block tile 64(M) x 128(N), K-stage 32, double-buffered LDS filled by TDM.
// 8 waves in a 2x4 grid; each wave register-blocks a 32x32 C tile:
// per k-step of 4 -> 2 A-frags + 2 B-frags feed 4 V_WMMA_F32_16X16X4_F32.
// Each wave DMAs its own 1/8 slice of the A and B tiles (no divergent issue),
// waits its own TENSORcnt, then the workgroup barrier publishes the buffer.
// ---------------------------------------------------------------------------
__global__ void __launch_bounds__(256)
k_encoder(const float* __restrict__ X, const float* __restrict__ Wenc,
          const float* __restrict__ benc, float* __restrict__ post) {
    __shared__ float As[2][BM][PAD_ROW];    // 16.9 KB
    __shared__ float Bs[2][BN][PAD_ROW];    // 33.8 KB

    const int tid   = threadIdx.x;
    const int tileN = blockIdx.x * BN;
    const int tileM = blockIdx.y * BM;
    const int wave  = tid >> 5, lane = tid & 31;
    const int wm    = wave >> 2, wn = wave & 3;      // 2 x 4 wave grid
    const int half  = lane >> 4, l15 = lane & 15;

    // per-wave TDM slices: A = 8 rows, B = 16 rows
    const int aRow = wave * (BM / 8);
    const int bRow = wave * (BN / 8);
    const unsigned long long aBase = (unsigned long long)(uintptr_t)X
        + 4ull * (unsigned long long)(tileM + aRow) * D_MODEL;
    const unsigned long long bBase = (unsigned long long)(uintptr_t)Wenc
        + 4ull * (unsigned long long)(tileN + bRow) * D_MODEL;
    const unsigned aLds[2] = { lds_byte_off(&As[0][aRow][0]), lds_byte_off(&As[1][aRow][0]) };
    const unsigned bLds[2] = { lds_byte_off(&Bs[0][bRow][0]), lds_byte_off(&Bs[1][bRow][0]) };

    // prologue: DMA stage 0 into buffer 0
    tdm_load_2d(aLds[0], aBase, D_MODEL, BATCH,  TK, BM / 8, D_MODEL);
    tdm_load_2d(bLds[0], bBase, D_MODEL, D_DICT, TK, BN / 8, D_MODEL);
    __builtin_amdgcn_s_wait_tensorcnt(0);
    __syncthreads();

    v8f c00 = {}, c01 = {}, c10 = {}, c11 = {};
    const int ar0 = wm * 32 + l15, ar1 = ar0 + 16;
    const int br0 = wn * 32 + l15, br1 = br0 + 16;

    for (int s = 0; s < KSTAGES; ++s) {
        const int cur = s & 1;
        if (s + 1 < KSTAGES) {          // prefetch next stage into other buffer
            const unsigned long long koff = 4ull * (unsigned)((s + 1) * TK);
            tdm_load_2d(aLds[cur ^ 1], aBase + koff, D_MODEL, BATCH,  TK, BM / 8, D_MODEL);
            tdm_load_2d(bLds[cur ^ 1], bBase + koff, D_MODEL, D_DICT, TK, BN / 8, D_MODEL);
        }
#pragma unroll
        for (int ks = 0; ks < TK / 4; ++ks) {
            const int kb = ks * 4 + 2 * half;   // A/B frag: VGPR0=K{0|2}, VGPR1=K{1|3}
            v2f a0, a1, b0, b1;
            a0.x = As[cur][ar0][kb]; a0.y = As[cur][ar0][kb + 1];
            a1.x = As[cur][ar1][kb]; a1.y = As[cur][ar1][kb + 1];
            b0.x = Bs[cur][br0][kb]; b0.y = Bs[cur][br0][kb + 1];
            b1.x = Bs[cur][br1][kb]; b1.y = Bs[cur][br1][kb + 1];
            c00 = __builtin_amdgcn_wmma_f32_16x16x4_f32(false, a0, false, b0, (short)0, c00, false, false);
            c01 = __builtin_amdgcn_wmma_f32_16x16x4_f32(false, a0, false, b1, (short)0, c01, false, false);
            c10 = __builtin_amdgcn_wmma_f32_16x16x4_f32(false, a1, false, b0, (short)0, c10, false, false);
            c11 = __builtin_amdgcn_wmma_f32_16x16x4_f32(false, a1, false, b1, (short)0, c11, false, false);
        }
        __builtin_amdgcn_s_wait_tensorcnt(0);   // next-stage DMA (issued above) landed
        __syncthreads();                        // publish buffer / retire reads
    }

    // epilogue: bias + relu, scatter the four 16x16 C tiles
    const int col0 = tileN + wn * 32 + l15;
    const int col1 = col0 + 16;
    const float bv0 = benc[col0], bv1 = benc[col1];
    const int row0 = tileM + wm * 32 + half * 8;   // VGPR i: M=i / M=8+i
    const int row1 = row0 + 16;
#pragma unroll
    for (int i = 0; i < 8; ++i) {
        float v;
        v = c00[i] + bv0; post[(size_t)(row0 + i) * D_DICT + col0] = v > 0.f ? v : 0.f;
        v = c01[i] + bv1; post[(size_t)(row0 + i) * D_DICT + col1] = v > 0.f ? v : 0.f;
        v = c10[i] + bv0; post[(size_t)(row1 + i) * D_DICT + col0] = v > 0.f ? v : 0.f;
        v = c11[i] + bv1; post[(size_t)(row1 + i) * D_DICT + col1] = v > 0.f ? v : 0.f;
    }
}

// ---------------------------------------------------------------------------
// Kernel 3: per-row exact top-K by 2-level radix select on float bits
// (relu output >= 0 => uint-monotonic). Row staged once in 192KB LDS, then
// exponent histogram -> mantissa refine -> 16-bit prefix threshold, masked
// in-place write (exactly K kept, ties capped), compaction of (idx,val).
// ---------------------------------------------------------------------------
__global__ void __launch_bounds__(512)
k_topk(float* __restrict__ latents, float* __restrict__ cvals,
       int* __restrict__ cidx, int* __restrict__ ccnt) {
    __shared__ float    srow[D_DICT];       // 192 KB row stage
    __shared__ unsigned hist[256];
    __shared__ int      s_a, s_b, s_take, s_cnt;

    const int row = blockIdx.x, tid = threadIdx.x;
    float* lrow = latents + (size_t)row * D_DICT;

    for (int i = tid; i < D_DICT; i += 512) srow[i] = lrow[i];
    if (tid < 256) hist[tid] = 0u;
    __syncthreads();

    // pass 1: exponent (bits[30:23]; sign bit is 0)
    for (int i = tid; i < D_DICT; i += 512) {
        unsigned b = __float_as_uint(srow[i]);
        atomicAdd(&hist[b >> 23], 1u);
    }
    __syncthreads();
    if (tid == 0) {
        unsigned cum = 0; int e = 255;
        for (; e > 0; --e) { unsigned h = hist[e]; if (cum + h >= TOPK) break; cum += h; }
        s_a = e;            // e*  (threshold exponent bucket)
        s_b = (int)cum;     // count strictly above e*
    }
    __syncthreads();
    const int estar = s_a, cAbove = s_b;
    if (tid < 256) hist[tid] = 0u;
    __syncthreads();

    // pass 2: top-8 mantissa bits within exponent bucket e*
    for (int i = tid; i < D_DICT; i += 512) {
        unsigned b = __float_as_uint(srow[i]);
        if ((int)(b >> 23) == estar) atomicAdd(&hist[(b >> 15) & 255], 1u);
    }
    __syncthreads();
    if (tid == 0) {
        unsigned cum = 0; int m = 255;
        for (; m > 0; --m) {
            unsigned h = hist[m];
            if ((unsigned)cAbove + cum + h >= TOPK) break;
            cum += h;
        }
        s_a = (estar << 8) | m;          // 16-bit prefix threshold
        s_b = cAbove + (int)cum;         // G = strictly greater than prefix
        s_take = 0; s_cnt = 0;
    }
    __syncthreads();
    const int thr16 = s_a;
    const int R     = TOPK - s_b;        // ties to keep at the threshold prefix

    // pass 3: masked in-place write + compaction
    for (int i = tid; i < D_DICT; i += 512) {
        float v = srow[i];
        int p = (int)(__float_as_uint(v) >> 15);
        bool keep = false;
        if (p > thr16) keep = true;
        else if (p == thr16) keep = (atomicAdd(&s_take, 1) < R);
        lrow[i] = keep ? v : 0.0f;
        if (keep) {
            int slot = atomicAdd(&s_cnt, 1);
            cvals[(size_t)row * KCAP + slot] = v;
            cidx [(size_t)row * KCAP + slot] = i;
        }
    }
    __syncthreads();
    if (tid == 0) ccnt[row] = s_cnt;     // == TOPK
}

// ---------------------------------------------------------------------------
// Kernel 4: sparse decode  x_hat[b,:] = sum_{j<K} v_j * WdecT[d_j, :]
// One block per row; active list staged in LDS; WdecT rows served from L2.
// ---------------------------------------------------------------------------
__global__ void __launch_bounds__(256)
k_decode(const float* __restrict__ WdecT, const float* __restrict__ cvals,
         const int* __restrict__ cidx, const int* __restrict__ ccnt,
         float* __restrict__ xhat) {
    __shared__ float sv[KCAP];
    __shared__ int   si[KCAP];
    const int row = blockIdx.x, tid = threadIdx.x;
    const int cnt = ccnt[row];
    for (int i = tid; i < cnt; i += 256) {
        sv[i] = cvals[(size_t)row * KCAP + i];
        si[i] = cidx [(size_t)row * KCAP + i];
    }
    __syncthreads();
    float a0 = 0.f, a1 = 0.f, a2 = 0.f;
    for (int j = 0; j < cnt; ++j) {
        const float v = sv[j];
        const float* wr = WdecT + (size_t)si[j] * D_MODEL;
        a0 = fmaf(v, wr[tid      ], a0);
        a1 = fmaf(v, wr[tid + 256], a1);
        a2 = fmaf(v, wr[tid + 512], a2);
    }
    float* out = xhat + (size_t)row * D_MODEL;
    out[tid]       = a0;
    out[tid + 256] = a1;
    out[tid + 512] = a2;
}

// ---------------------------------------------------------------------------
extern "C" void kernel_launch(void* const* d_in, const int* in_sizes, int n_in,
                              void* d_out, int out_size, void* d_ws, size_t ws_size,
                              hipStream_t stream) {
    const float* x    = (const float*)d_in[0];
    const float* Wenc = (const float*)d_in[1];
    const float* benc = (const float*)d_in[2];
    const float* Wdec = (const float*)d_in[3];

    float* xhat    = (float*)d_out;                          // [4096 x 768]
    float* latents = xhat + (size_t)BATCH * D_MODEL;         // [4096 x 49152]

    // workspace layout: WdecT | cvals | cidx | ccnt  (~168 MB total)
    float* WdecT = (float*)d_ws;
    float* cvals = WdecT + (size_t)D_DICT * D_MODEL;
    int*   cidx  = (int*)(cvals + (size_t)BATCH * KCAP);
    int*   ccnt  = cidx + (size_t)BATCH * KCAP;

    k_transpose<<<dim3(D_DICT / 32, D_MODEL / 32), dim3(32, 8), 0, stream>>>(Wdec, WdecT);
    k_encoder <<<dim3(D_DICT / BN, BATCH / BM), 256, 0, stream>>>(x, Wenc, benc, latents);
    k_topk    <<<BATCH, 512, 0, stream>>>(latents, cvals, cidx, ccnt);
    k_decode  <<<BATCH, 256, 0, stream>>>(WdecT, cvals, cidx, ccnt, xhat);
}